// AttRnnBlk_41257455845893
// MI455X (gfx1250) — compile-verified
//
#include <hip/hip_runtime.h>
#include <math.h>

// Problem constants (fixed by setup_inputs in the reference)
#define BB   64      // batch
#define SS   512     // sequence
#define DD   512     // feature dim
#define HH   512     // hidden dim
#define NHEAD 4      // heads
#define ND   (NHEAD*DD)   // 2048
#define TSTEP 64     // outp
#define EPSV 1e-6f
#define NEPS 1e-12f

typedef __attribute__((ext_vector_type(2))) float v2f;
typedef __attribute__((ext_vector_type(8))) float v8f;

// D = A(16x4, f32) x B(4x16, f32) + C(16x16, f32), full-precision WMMA
static __device__ __forceinline__ v8f wmma4(v2f a, v2f b, v8f c) {
  return __builtin_amdgcn_wmma_f32_16x16x4_f32(false, a, false, b, (short)0, c,
                                               false, false);
}

static __device__ __forceinline__ float sigmoidf_(float x) {
  return 1.0f / (1.0f + __expf(-x));
}

// ---------------------------------------------------------------------------
// K0a: copy h0/c0 into workspace (h is ping-ponged across steps)
__global__ void init_state_kernel(const float* __restrict__ h0,
                                  const float* __restrict__ c0,
                                  float* __restrict__ h, float* __restrict__ c) {
  int i = blockIdx.x * blockDim.x + threadIdx.x;
  if (i < BB * HH) { h[i] = h0[i]; c[i] = c0[i]; }
}

// K0b: x_sqrt[b,s] = ||x[b,s,:]||  (one wave32 per row)
__global__ void xsqrt_kernel(const float* __restrict__ x, float* __restrict__ xsq) {
  int row  = (blockIdx.x * blockDim.x + threadIdx.x) >> 5;  // b*S+s
  int lane = threadIdx.x & 31;
  const float* xr = x + (size_t)row * DD;
  float s = 0.f;
  for (int d = lane; d < DD; d += 32) { float t = xr[d]; s += t * t; }
  for (int off = 16; off; off >>= 1) s += __shfl_down(s, off);
  if (lane == 0) xsq[row] = sqrtf(s);
}

// ---------------------------------------------------------------------------
// K1: key[b, n*D+d] = sum_k h[b,k] * densa_W[(n*D+d)*H + k] + densa_b[n*D+d]
// GEMM M=64, N=2048, K=512. One wave per 16x16 tile; 512 waves total.
__global__ void key_gemm_kernel(const float* __restrict__ h,
                                const float* __restrict__ W,   // [2048,512] row-major (=Wt)
                                const float* __restrict__ bias,
                                float* __restrict__ key) {
  int wid  = (blockIdx.x * blockDim.x + threadIdx.x) >> 5;
  int lane = threadIdx.x & 31;
  int mt = wid >> 7;            // 0..3
  int nt = wid & 127;           // 0..127
  int m0 = mt * 16, n0 = nt * 16;
  int r  = lane & 15;           // A-row / B-col index for this lane
  int kg = lane >> 4;           // K-group: lanes 16-31 hold K+2,K+3

  v8f acc0 = {}, acc1 = {};
#pragma unroll 4
  for (int k = 0; k < HH; k += 8) {
    int ka = k + 2 * kg;
    v2f a0 = *(const v2f*)(h + (size_t)(m0 + r) * HH + ka);
    v2f b0 = *(const v2f*)(W + (size_t)(n0 + r) * HH + ka);
    acc0 = wmma4(a0, b0, acc0);
    v2f a1 = *(const v2f*)(h + (size_t)(m0 + r) * HH + ka + 4);
    v2f b1 = *(const v2f*)(W + (size_t)(n0 + r) * HH + ka + 4);
    acc1 = wmma4(a1, b1, acc1);
  }
  int colN = lane & 15, mbase = (lane >> 4) * 8;
#pragma unroll
  for (int g = 0; g < 8; ++g) {
    int m = m0 + mbase + g, n = n0 + colN;
    key[(size_t)m * ND + n] = acc0[g] + acc1[g] + bias[n];
  }
}

// ---------------------------------------------------------------------------
// K2: fused attention per batch b (one 256-thread block = 8 waves per b):
//   k_norm -> inner = key.x^T (WMMA, M=4 padded to 16) -> w normalize (LDS)
//   -> val = w.x (WMMA from LDS)
__global__ void attn_kernel(const float* __restrict__ x,
                            const float* __restrict__ key,
                            const float* __restrict__ xsq,
                            float* __restrict__ val) {
  __shared__ float wlds[NHEAD][SS];   // 8 KB: normalized attention weights
  __shared__ float partial[8];
  __shared__ float p2[8][NHEAD];
  __shared__ float knorm[NHEAD];
  __shared__ float wnrm[NHEAD];

  int b    = blockIdx.x;
  int tid  = threadIdx.x;
  int wv   = tid >> 5;
  int lane = tid & 31;
  int colN = lane & 15;
  int kg   = lane >> 4;
  int mbase = kg * 8;

  const float* kb = key + (size_t)b * ND;      // key[b, n*512+d]
  const float* xb = x   + (size_t)b * SS * DD; // x[b, s, d]

  // --- k_norm: wave wv reduces half a head row ---
  {
    int n = wv & 3, half = wv >> 2;
    const float* krow = kb + n * DD + half * 256;
    float s = 0.f;
    for (int d = lane; d < 256; d += 32) { float t = krow[d]; s += t * t; }
    for (int off = 16; off; off >>= 1) s += __shfl_down(s, off);
    if (lane == 0) partial[wv] = s;
  }
  __syncthreads();
  if (tid < NHEAD) knorm[tid] = sqrtf(partial[tid] + partial[tid + 4]);
  __syncthreads();

  // --- inner + raw w: wave wv owns 4 S-tiles of 16 ---
  float sq[NHEAD] = {0.f, 0.f, 0.f, 0.f};
  for (int t4 = 0; t4 < 4; ++t4) {
    int s0 = (wv * 4 + t4) * 16;
    v8f acc = {};
#pragma unroll 4
    for (int k = 0; k < DD; k += 4) {
      int ka = k + 2 * kg;
      v2f a = {0.f, 0.f};
      if (colN < NHEAD) a = *(const v2f*)(kb + (size_t)colN * DD + ka);
      v2f bv = *(const v2f*)(xb + (size_t)(s0 + colN) * DD + ka); // B[k,n]=x[b,s0+n,k]
      acc = wmma4(a, bv, acc);
    }
#pragma unroll
    for (int g = 0; g < 8; ++g) {
      int m = mbase + g;                 // head index (valid rows 0..3, lanes 0-15)
      if (m < NHEAD) {
        int sc = s0 + colN;
        float wr = acc[g] / (knorm[m] + xsq[(size_t)b * SS + sc] + EPSV);
        wlds[m][sc] = wr;
        sq[m] += wr * wr;
      }
    }
  }
  // reduce sum-of-squares per head (lanes 16-31 contribute zeros)
#pragma unroll
  for (int off = 16; off; off >>= 1)
    for (int n = 0; n < NHEAD; ++n) sq[n] += __shfl_down(sq[n], off);
  if (lane == 0)
    for (int n = 0; n < NHEAD; ++n) p2[wv][n] = sq[n];
  __syncthreads();
  if (tid < NHEAD) {
    float t = 0.f;
    for (int w = 0; w < 8; ++w) t += p2[w][tid];
    wnrm[tid] = 1.0f / fmaxf(sqrtf(t), NEPS);
  }
  __syncthreads();
  for (int i = tid; i < NHEAD * SS; i += 256) wlds[i >> 9][i & 511] *= wnrm[i >> 9];
  __syncthreads();

  // --- val[b,n,d] = sum_s w[n,s] x[b,s,d]: A from LDS, B row-major from x ---
  for (int t4 = 0; t4 < 4; ++t4) {
    int d0 = (wv * 4 + t4) * 16;
    v8f acc = {};
#pragma unroll 4
    for (int k = 0; k < SS; k += 4) {
      int ka = k + 2 * kg;
      v2f a = {0.f, 0.f};
      if (colN < NHEAD) { a.x = wlds[colN][ka]; a.y = wlds[colN][ka + 1]; }
      v2f bv;
      bv.x = xb[(size_t)ka * DD + d0 + colN];        // B[k,n] = x[b,k,d0+n]
      bv.y = xb[(size_t)(ka + 1) * DD + d0 + colN];
      acc = wmma4(a, bv, acc);
    }
#pragma unroll
    for (int g = 0; g < 8; ++g) {
      int m = mbase + g;
      if (m < NHEAD) val[(size_t)b * ND + m * DD + d0 + colN] = acc[g];
    }
  }
}

// ---------------------------------------------------------------------------
// K3: v[b,d] = tanh( sum_j val[b,j] * sumary_W[d*2048 + j] + sumary_b[d] )
// GEMM M=64, N=512, K=2048. 128 wave-tiles.
__global__ void sumary_gemm_kernel(const float* __restrict__ val,
                                   const float* __restrict__ W,   // [512,2048]
                                   const float* __restrict__ bias,
                                   float* __restrict__ v) {
  int wid  = (blockIdx.x * blockDim.x + threadIdx.x) >> 5;
  int lane = threadIdx.x & 31;
  int mt = wid >> 5;           // 0..3
  int nt = wid & 31;           // 0..31
  int m0 = mt * 16, n0 = nt * 16;
  int r = lane & 15, kg = lane >> 4;

  v8f acc0 = {}, acc1 = {};
#pragma unroll 4
  for (int k = 0; k < ND; k += 8) {
    int ka = k + 2 * kg;
    v2f a0 = *(const v2f*)(val + (size_t)(m0 + r) * ND + ka);
    v2f b0 = *(const v2f*)(W   + (size_t)(n0 + r) * ND + ka);
    acc0 = wmma4(a0, b0, acc0);
    v2f a1 = *(const v2f*)(val + (size_t)(m0 + r) * ND + ka + 4);
    v2f b1 = *(const v2f*)(W   + (size_t)(n0 + r) * ND + ka + 4);
    acc1 = wmma4(a1, b1, acc1);
  }
  int colN = lane & 15, mbase = (lane >> 4) * 8;
#pragma unroll
  for (int g = 0; g < 8; ++g) {
    int m = m0 + mbase + g, n = n0 + colN;
    v[(size_t)m * DD + n] = tanhf(acc0[g] + acc1[g] + bias[n]);
  }
}

// ---------------------------------------------------------------------------
// K4: fused LSTM cell. Each wave owns (16 batch rows) x (16 hidden cols) and
// accumulates all four gate quadrants (i,f,g,o) from both W_ih and W_hh,
// then applies the cell update and writes h_new, c_new, and hs[b,t,:].
__global__ void lstm_kernel(const float* __restrict__ v,
                            const float* __restrict__ hin,
                            const float* __restrict__ Wih,  // [2048,512]
                            const float* __restrict__ Whh,  // [2048,512]
                            const float* __restrict__ bih,
                            const float* __restrict__ bhh,
                            const float* __restrict__ cin,
                            float* __restrict__ hout,
                            float* __restrict__ cout,
                            float* __restrict__ hs_out, int t) {
  int wid  = (blockIdx.x * blockDim.x + threadIdx.x) >> 5;
  int lane = threadIdx.x & 31;
  int mt = wid >> 5, nt = wid & 31;
  int m0 = mt * 16, n0 = nt * 16;
  int r = lane & 15, kg = lane >> 4;

  v8f ac[4] = {{}, {}, {}, {}};
#pragma unroll 2
  for (int k = 0; k < HH; k += 4) {
    int ka = k + 2 * kg;
    v2f av = *(const v2f*)(v   + (size_t)(m0 + r) * DD + ka);
    v2f ah = *(const v2f*)(hin + (size_t)(m0 + r) * HH + ka);
#pragma unroll
    for (int q = 0; q < 4; ++q) {
      int row = q * HH + n0 + r;
      v2f b1 = *(const v2f*)(Wih + (size_t)row * DD + ka);
      ac[q] = wmma4(av, b1, ac[q]);
      v2f b2 = *(const v2f*)(Whh + (size_t)row * HH + ka);
      ac[q] = wmma4(ah, b2, ac[q]);
    }
  }
  int colN = lane & 15, mbase = (lane >> 4) * 8;
#pragma unroll
  for (int g8 = 0; g8 < 8; ++g8) {
    int m = m0 + mbase + g8;   // batch row
    int n = n0 + colN;         // hidden col
    float gi = sigmoidf_(ac[0][g8] + bih[n]          + bhh[n]);
    float gf = sigmoidf_(ac[1][g8] + bih[HH + n]     + bhh[HH + n]);
    float gg = tanhf   (ac[2][g8] + bih[2 * HH + n] + bhh[2 * HH + n]);
    float go = sigmoidf_(ac[3][g8] + bih[3 * HH + n] + bhh[3 * HH + n]);
    float cn = gf * cin[(size_t)m * HH + n] + gi * gg;
    float hn = go * tanhf(cn);
    cout[(size_t)m * HH + n] = cn;
    hout[(size_t)m * HH + n] = hn;
    hs_out[((size_t)m * TSTEP + t) * HH + n] = hn;
  }
}

// ---------------------------------------------------------------------------
extern "C" void kernel_launch(void* const* d_in, const int* in_sizes, int n_in,
                              void* d_out, int out_size, void* d_ws, size_t ws_size,
                              hipStream_t stream) {
  (void)in_sizes; (void)n_in; (void)out_size; (void)ws_size;
  const float* x        = (const float*)d_in[0];
  const float* h0       = (const float*)d_in[1];
  const float* c0       = (const float*)d_in[2];
  const float* densa_W  = (const float*)d_in[3];
  const float* densa_b  = (const float*)d_in[4];
  const float* sumary_W = (const float*)d_in[5];
  const float* sumary_b = (const float*)d_in[6];
  const float* W_ih     = (const float*)d_in[7];
  const float* W_hh     = (const float*)d_in[8];
  const float* b_ih     = (const float*)d_in[9];
  const float* b_hh     = (const float*)d_in[10];
  float* out = (float*)d_out;

  float* ws  = (float*)d_ws;
  float* hA  = ws;                 // 32768
  float* hB  = hA + BB * HH;       // 32768
  float* c   = hB + BB * HH;       // 32768
  float* key = c  + BB * HH;       // 131072
  float* val = key + BB * ND;      // 131072
  float* vb  = val + BB * ND;      // 32768
  float* xsq = vb + BB * DD;       // 32768

  init_state_kernel<<<(BB * HH + 255) / 256, 256, 0, stream>>>(h0, c0, hA, c);
  xsqrt_kernel<<<(BB * SS) / 8, 256, 0, stream>>>(x, xsq);  // 8 waves/block

  for (int t = 0; t < TSTEP; ++t) {
    float* hcur = (t & 1) ? hB : hA;
    float* hnxt = (t & 1) ? hA : hB;
    key_gemm_kernel<<<64, 256, 0, stream>>>(hcur, densa_W, densa_b, key);
    attn_kernel<<<BB, 256, 0, stream>>>(x, key, xsq, val);
    sumary_gemm_kernel<<<16, 256, 0, stream>>>(val, sumary_W, sumary_b, vb);
    lstm_kernel<<<16, 256, 0, stream>>>(vb, hcur, W_ih, W_hh, b_ih, b_hh,
                                        c, hnxt, c, out, t);
  }
}